// CoFiSet_71966472011947
// MI455X (gfx1250) — compile-verified
//
#include <hip/hip_runtime.h>

typedef __attribute__((ext_vector_type(2))) float v2f;
typedef __attribute__((ext_vector_type(8))) float v8f;

#define S_POS 20
#define DIM   64

__device__ __forceinline__ float pick8(v8f a, int i) {
  float r = a[0];
  if (i == 1) r = a[1];
  if (i == 2) r = a[2];
  if (i == 3) r = a[3];
  if (i == 4) r = a[4];
  if (i == 5) r = a[5];
  if (i == 6) r = a[6];
  if (i == 7) r = a[7];
  return r;
}

// One wave32 handles 16 consecutive batch elements.
// A (16x64 f32) = gathered user rows; B (64x16) = transposed item rows.
// dot(u_b, v_b) = diag of A*B^T computed with chained v_wmma_f32_16x16x4_f32.
__global__ __launch_bounds__(256) void cofiset_score_kernel(
    const float* __restrict__ U, const float* __restrict__ V,
    const float* __restrict__ bi, const int* __restrict__ user_id,
    const int* __restrict__ pos_item_id, const int* __restrict__ neg_item_id,
    float* __restrict__ out, int B) {
  const int lane = threadIdx.x & 31;
  const int wave = blockIdx.x * (blockDim.x >> 5) + (threadIdx.x >> 5);
  const int b0   = wave * 16;
  if (b0 >= B) return;                 // wave-uniform

  const int n       = lane & 15;      // which of the 16 batch rows this lane serves
  const int hi      = lane >> 4;      // K-split half (see ISA 16x4/4x16 layouts)
  const int colBase = 2 * hi;         // float offset within each 4-wide K chunk

  int bb = b0 + n;                    // clamped gather index for partial tiles
  if (bb >= B) bb = B - 1;

  // ---- Phase 1: stream + sum the 20 positive V rows into WMMA-B layout ----
  v2f bsum[16];
#pragma unroll
  for (int k = 0; k < 16; ++k) { v2f z = {0.f, 0.f}; bsum[k] = z; }

  const int* prow = pos_item_id + (size_t)bb * S_POS;
  for (int s = 0; s < S_POS; ++s) {
    const float* vr = V + (size_t)prow[s] * DIM + colBase;
#pragma unroll
    for (int k = 0; k < 16; ++k) {
      v2f t = *(const v2f*)(vr + 4 * k);
      bsum[k] += t;
    }
  }

  // ---- Phase 2: gather user row into WMMA-A layout (reused by both chains) ----
  v2f arow[16];
  {
    const float* ur = U + (size_t)user_id[bb] * DIM + colBase;
#pragma unroll
    for (int k = 0; k < 16; ++k) arow[k] = *(const v2f*)(ur + 4 * k);
  }

  // ---- Phase 3: positive dots, 16 chained f32 WMMAs over K=64 ----
  v8f accP = {};
#pragma unroll
  for (int k = 0; k < 16; ++k)
    accP = __builtin_amdgcn_wmma_f32_16x16x4_f32(
        false, arow[k], false, bsum[k], (short)0, accP, false, false);

  // ---- Phase 4: negative dots ----
  v8f accN = {};
  {
    const float* vr = V + (size_t)neg_item_id[bb] * DIM + colBase;
#pragma unroll
    for (int k = 0; k < 16; ++k) {
      v2f t = *(const v2f*)(vr + 4 * k);
      accN = __builtin_amdgcn_wmma_f32_16x16x4_f32(
          false, arow[k], false, t, (short)0, accN, false, false);
    }
  }

  // ---- Phase 5: diagonal extraction + biases + store (divergent, post-WMMA) ----
  // diag element n: VGPR n, lane n (n<8); VGPR n-8, lane n+16 (n>=8)
  const bool active = (hi == 0) ? (n < 8) : (n >= 8);
  if (active && (b0 + n) < B) {
    const int b  = b0 + n;
    const float dp = pick8(accP, n & 7);
    const float dn = pick8(accN, n & 7);
    float bs = 0.f;
    const int* pr = pos_item_id + (size_t)b * S_POS;
#pragma unroll
    for (int s = 0; s < S_POS; ++s) bs += bi[pr[s]];
    out[b]     = (dp + bs) * (1.0f / S_POS);
    out[B + b] = dn + bi[neg_item_id[b]];
  }
}

extern "C" void kernel_launch(void* const* d_in, const int* in_sizes, int n_in,
                              void* d_out, int out_size, void* d_ws, size_t ws_size,
                              hipStream_t stream) {
  const float* U           = (const float*)d_in[0];
  const float* V           = (const float*)d_in[1];
  const float* bi          = (const float*)d_in[2];
  const int*   user_id     = (const int*)d_in[3];
  const int*   pos_item_id = (const int*)d_in[4];
  const int*   neg_item_id = (const int*)d_in[5];
  float*       out         = (float*)d_out;

  const int B = in_sizes[3];                 // batch size from user_id
  const int waves = (B + 15) / 16;
  const int threads = 256;                   // 8 wave32 per block
  const int wavesPerBlock = threads / 32;
  const int blocks = (waves + wavesPerBlock - 1) / wavesPerBlock;

  cofiset_score_kernel<<<blocks, threads, 0, stream>>>(
      U, V, bi, user_id, pos_item_id, neg_item_id, out, B);
}